// UNET_block_8478265442556
// MI455X (gfx1250) — compile-verified
//
#include <hip/hip_runtime.h>
#include <math.h>

// ---------------------------------------------------------------------------
// Involution residual block for MI455X (gfx1250, wave32, WMMA).
// All 1x1 convs -> V_WMMA_F32_16X16X4_F32 GEMMs with fused affine/act/residual.
// Involution apply -> LDS-tiled VALU kernel with fused affine+GELU.
//
// Roofline: ~2 GFLOP vs ~250 MB of intermediate traffic -> near the 23.3 TB/s
// memory roof, so fp32 WMMA (exact vs reference) is the right precision; all
// intermediates are L2-resident (192 MB).
//
// All BN/bias epilogues are prefolded once into per-channel (scale,shift)
// float2 tables by a one-block prep kernel, so GEMM epilogues are a single
// b64 load + FMA per element -- no per-element rsqrt chains / pointer tests.
// ---------------------------------------------------------------------------

typedef float v2f __attribute__((ext_vector_type(2)));
typedef float v8f __attribute__((ext_vector_type(8)));

#define IMG     128
#define HW      16384          // IMG*IMG
#define NT      1024           // HW / 16 column tiles per batch image
#define NT_LOG  10
#define NBATCH  4

// (scale,shift) table offsets inside the params block (units of float2)
#define AFF_R   0              // reduce BN            : 16
#define AFF1    16             // post-involution BN   : 64
#define AFF2    80             // conv2 BN             : 64
#define AFFM    144            // mapping BN (+bias)   : 64
#define AFFS    208            // span bias as (1,bs)  : 208 (196 padded)
#define PARAMS_FLOATS 1024     // 4 KB reserved at head of workspace

__device__ __forceinline__ float gelu_erf(float x) {
    // torch nn.GELU default: 0.5*x*(1+erf(x/sqrt(2)))
    return 0.5f * x * (1.0f + erff(x * 0.70710678118654752f));
}

// ---------------------------------------------------------------------------
// One-block prep: fold every BN (+ pre-BN bias) into (scale, shift) pairs.
//   y = BN(v + bias) = v*s + (b - m*s + bias*s),  s = g*rsqrt(var+eps)
// ---------------------------------------------------------------------------
__global__ __launch_bounds__(256)
void bn_prep(const float* __restrict__ gr, const float* __restrict__ br,
             const float* __restrict__ mr, const float* __restrict__ vr,
             const float* __restrict__ g1, const float* __restrict__ b1,
             const float* __restrict__ m1, const float* __restrict__ v1,
             const float* __restrict__ g2, const float* __restrict__ b2,
             const float* __restrict__ m2, const float* __restrict__ v2,
             const float* __restrict__ gm, const float* __restrict__ bm,
             const float* __restrict__ mm, const float* __restrict__ vm,
             const float* __restrict__ bmap, const float* __restrict__ bs,
             v2f* __restrict__ P)
{
    const int i = threadIdx.x;
    if (i < 16) {
        const float s = gr[i] * rsqrtf(vr[i] + 1e-5f);
        P[AFF_R + i] = (v2f){s, br[i] - mr[i] * s};
    }
    if (i < 64) {
        float s = g1[i] * rsqrtf(v1[i] + 1e-5f);
        P[AFF1 + i] = (v2f){s, b1[i] - m1[i] * s};
        s = g2[i] * rsqrtf(v2[i] + 1e-5f);
        P[AFF2 + i] = (v2f){s, b2[i] - m2[i] * s};
        s = gm[i] * rsqrtf(vm[i] + 1e-5f);
        P[AFFM + i] = (v2f){s, bm[i] - mm[i] * s + bmap[i] * s};
    }
    if (i < 208) {
        P[AFFS + i] = (v2f){1.f, (i < 196) ? bs[i] : 0.f};
    }
}

// ---------------------------------------------------------------------------
// Wave-per-tile fp32 WMMA GEMM:  Y[b,m,n] = ACT( (W@X)*aff.x + aff.y [+ addend] )
// X layout [B, K, HW], Y layout [B, M, HW].  Each wave computes a 16(M)x16(N)
// tile, accumulating over K in steps of 4 with v_wmma_f32_16x16x4_f32.
//
// Fragment layouts (ISA 7.12.2, 32-bit 16x4 A / 4x16 B / 16x16 C):
//   A: lane l holds row m = l&15, K pair {2*(l>>4), 2*(l>>4)+1} in {a.x, a.y}
//   B: lane l holds col n = l&15, same K pair in {b.x, b.y}
//   D: element v of lane l is (m = v + 8*(l>>4), n = l&15)
//
// GUARD=true only for the span conv (M=196, 13 padded row tiles): A rows load
// unconditionally from a clamped valid row, then branch-free multiply-mask.
// All K/4 A and B fragments are register-staged so the scheduler emits one
// load clause, then a back-to-back WMMA chain with staggered loadcnt waits.
// ACT: 0 = none, 1 = ReLU, 2 = erf-GELU
// ---------------------------------------------------------------------------
template<int K, bool GUARD, bool AFF, bool ADD, int ACT>
__global__ __launch_bounds__(256)
void wmma_gemm_f32(const float* __restrict__ X, const float* __restrict__ W,
                   const v2f* __restrict__ aff, const float* __restrict__ addend,
                   float* __restrict__ Y, int M, int Mt, int total_waves)
{
    constexpr int KS = K / 4;

    const int lane = threadIdx.x & 31;
    const int gw   = blockIdx.x * (blockDim.x >> 5) + (threadIdx.x >> 5);
    if (gw >= total_waves) return;          // whole-wave exit: EXEC stays all-1s for WMMA

    const int mt = gw % Mt;
    const int t  = gw / Mt;
    const int nt = t & (NT - 1);
    const int b  = t >> NT_LOG;

    const int l15  = lane & 15;
    const int half = lane >> 4;
    const int n    = nt * 16 + l15;         // output column (pixel within image)
    const int mA   = mt * 16 + l15;         // A-fragment row

    // Clamped, always-valid row pointer -> unconditional (branch-free) loads.
    const int   mclamp = (GUARD && mA >= M) ? (M - 1) : mA;
    const float amask  = (GUARD && mA >= M) ? 0.f : 1.f;
    const float* wrow  = W + (size_t)mclamp * K;
    const float* xcol  = X + (size_t)b * K * HW + n;

    // Stage all fragments into registers (<= 16 v2f each), then WMMA chain.
    v2f afrag[KS];
    v2f bfrag[KS];
#pragma unroll
    for (int s = 0; s < KS; ++s) {
        const int kk = 4 * s + 2 * half;
        afrag[s] = *(const v2f*)(wrow + kk);            // b64 load, K-pair contiguous
        bfrag[s].x = xcol[(size_t)kk * HW];
        bfrag[s].y = xcol[(size_t)(kk + 1) * HW];
    }
    if (GUARD) {
#pragma unroll
        for (int s = 0; s < KS; ++s) {                  // branch-free zero of pad rows
            afrag[s].x *= amask;
            afrag[s].y *= amask;
        }
    }

    v8f acc = {0.f, 0.f, 0.f, 0.f, 0.f, 0.f, 0.f, 0.f};
#pragma unroll
    for (int s = 0; s < KS; ++s) {
        acc = __builtin_amdgcn_wmma_f32_16x16x4_f32(
            /*neg_a=*/false, afrag[s], /*neg_b=*/false, bfrag[s],
            /*c_mod=*/(short)0, acc, /*reuse_a=*/false, /*reuse_b=*/false);
    }

    // Lean fused epilogue: one b64 (scale,shift) load + FMA per element.
#pragma unroll
    for (int v = 0; v < 8; ++v) {
        const int mo = mt * 16 + v + 8 * half;
        if (GUARD && mo >= M) continue;
        float val = acc[v];
        if (AFF) {
            const v2f p = aff[mo];
            val = val * p.x + p.y;
        }
        const size_t oidx = ((size_t)(b * M + mo)) * HW + n;
        if (ADD) val += addend[oidx];
        if (ACT == 1)      val = fmaxf(val, 0.f);
        else if (ACT == 2) val = gelu_erf(val);
        Y[oidx] = val;
    }
}

// ---------------------------------------------------------------------------
// Involution apply + BN + GELU.
//   y[b, g*16+c, p] = gelu( aff1[c]( sum_k wk[b,g*49+k,p] * x1[b,g*16+c,p+off(k)] ) )
// Block = 16x16 pixel tile for one (batch, group); the 16-channel group with a
// 3-pixel halo is staged in LDS (16*22*24*4B = 33 KB of the 320 KB WGP LDS).
// Each thread owns one pixel and accumulates all 16 group channels in VGPRs;
// wk values are per-pixel-private so they stream straight from global/L2.
// ---------------------------------------------------------------------------
__global__ __launch_bounds__(256)
void involution_apply(const float* __restrict__ x1, const float* __restrict__ wk,
                      const v2f* __restrict__ aff1, float* __restrict__ y)
{
    __shared__ float tile[16][22][24];   // [c][h+halo][w+halo], padded stride vs bank conflicts

    const int tx = threadIdx.x & 15;
    const int ty = threadIdx.x >> 4;
    const int b  = blockIdx.z >> 2;
    const int g  = blockIdx.z & 3;
    const int h0 = blockIdx.y * 16;
    const int w0 = blockIdx.x * 16;

    // Cooperative halo load with zero padding (pad=3 each side).
    const float* xg = x1 + ((size_t)(b * 64 + g * 16)) * HW;
    for (int i = threadIdx.x; i < 16 * 22 * 22; i += 256) {
        const int c  = i / 484;
        const int r  = i - c * 484;
        const int hh = r / 22;
        const int ww = r - hh * 22;
        const int gh = h0 + hh - 3;
        const int gw = w0 + ww - 3;
        float vv = 0.f;
        if (gh >= 0 && gh < IMG && gw >= 0 && gw < IMG)
            vv = xg[(size_t)c * HW + gh * IMG + gw];
        tile[c][hh][ww] = vv;
    }
    __syncthreads();

    float acc[16];
#pragma unroll
    for (int c = 0; c < 16; ++c) acc[c] = 0.f;

    const int pix = (h0 + ty) * IMG + (w0 + tx);
    const float* wkp = wk + ((size_t)(b * 196 + g * 49)) * HW + pix;

    for (int k = 0; k < 49; ++k) {       // 7x7 dynamic kernel, nn.Unfold order
        const float wv = wkp[(size_t)k * HW];
        const int dh = k / 7;
        const int dw = k - dh * 7;
#pragma unroll
        for (int c = 0; c < 16; ++c)
            acc[c] += wv * tile[c][ty + dh][tx + dw];
    }

    // Fused prefolded BN + GELU on the way out.
    float* yp = y + ((size_t)(b * 64 + g * 16)) * HW + pix;
#pragma unroll
    for (int c = 0; c < 16; ++c) {
        const v2f p = aff1[g * 16 + c];
        yp[(size_t)c * HW] = gelu_erf(acc[c] * p.x + p.y);
    }
}

// ---------------------------------------------------------------------------
extern "C" void kernel_launch(void* const* d_in, const int* in_sizes, int n_in,
                              void* d_out, int out_size, void* d_ws, size_t ws_size,
                              hipStream_t stream)
{
    (void)in_sizes; (void)n_in; (void)out_size; (void)ws_size;

    const float* x    = (const float*)d_in[0];
    const float* w1   = (const float*)d_in[1];
    const float* wr   = (const float*)d_in[2];
    const float* gr   = (const float*)d_in[3];
    const float* br   = (const float*)d_in[4];
    const float* mr   = (const float*)d_in[5];
    const float* vr   = (const float*)d_in[6];
    const float* wsp  = (const float*)d_in[7];   // span weight "ws"
    const float* bs   = (const float*)d_in[8];
    const float* g1   = (const float*)d_in[9];
    const float* b1   = (const float*)d_in[10];
    const float* m1   = (const float*)d_in[11];
    const float* v1   = (const float*)d_in[12];
    const float* w2   = (const float*)d_in[13];
    const float* g2   = (const float*)d_in[14];
    const float* b2   = (const float*)d_in[15];
    const float* m2   = (const float*)d_in[16];
    const float* v2   = (const float*)d_in[17];
    const float* wm   = (const float*)d_in[18];
    const float* bmap = (const float*)d_in[19];
    const float* gm   = (const float*)d_in[20];
    const float* betam= (const float*)d_in[21];
    const float* mm   = (const float*)d_in[22];
    const float* vm   = (const float*)d_in[23];
    float* out = (float*)d_out;

    // Workspace layout (floats): PARAMS | X1 | SKIP | R | WK | X2  (~106 MB)
    float* Pf   = (float*)d_ws;                          // 1024 floats of (sc,sh) tables
    v2f*   P    = (v2f*)Pf;
    float* X1   = Pf   + PARAMS_FLOATS;                  // [4, 64,HW]  gelu(conv1)
    float* SKIP = X1   + (size_t)NBATCH * 64  * HW;      // [4, 64,HW]  bn(mapping)
    float* R    = SKIP + (size_t)NBATCH * 64  * HW;      // [4, 16,HW]  relu(bn(reduce))
    float* WK   = R    + (size_t)NBATCH * 16  * HW;      // [4,196,HW]  span kernels
    float* X2   = WK   + (size_t)NBATCH * 196 * HW;      // [4, 64,HW]  gelu(bn(involution))

    auto nblk = [](int waves) { return (waves + 7) / 8; };   // 8 wave32/block

    // 0) Fold all BN/bias epilogues into (scale,shift) tables.
    bn_prep<<<1, 256, 0, stream>>>(gr, br, mr, vr, g1, b1, m1, v1,
                                   g2, b2, m2, v2, gm, betam, mm, vm,
                                   bmap, bs, P);

    // 1) X1 = GELU(w1 @ x)                          M=64, K=32
    { const int Mt = 4, tw = Mt * NT * NBATCH;
      wmma_gemm_f32<32, false, false, false, 2><<<nblk(tw), 256, 0, stream>>>(
          x, w1, nullptr, nullptr, X1, 64, Mt, tw); }

    // 2) SKIP = BN(wm @ x + bmap)                   M=64, K=32
    { const int Mt = 4, tw = Mt * NT * NBATCH;
      wmma_gemm_f32<32, false, true, false, 0><<<nblk(tw), 256, 0, stream>>>(
          x, wm, P + AFFM, nullptr, SKIP, 64, Mt, tw); }

    // 3) R = ReLU(BN(wr @ X1))                      M=16, K=64
    { const int Mt = 1, tw = Mt * NT * NBATCH;
      wmma_gemm_f32<64, false, true, false, 1><<<nblk(tw), 256, 0, stream>>>(
          X1, wr, P + AFF_R, nullptr, R, 16, Mt, tw); }

    // 4) WK = wsp @ R + bs                          M=196 (13 padded tiles), K=16
    { const int Mt = 13, tw = Mt * NT * NBATCH;
      wmma_gemm_f32<16, true, true, false, 0><<<nblk(tw), 256, 0, stream>>>(
          R, wsp, P + AFFS, nullptr, WK, 196, Mt, tw); }

    // 5) X2 = GELU(BN(involution(X1, WK)))
    involution_apply<<<dim3(IMG / 16, IMG / 16, NBATCH * 4), 256, 0, stream>>>(
        X1, WK, P + AFF1, X2);

    // 6) out = GELU(BN(w2 @ X2) + SKIP)             M=64, K=64
    { const int Mt = 4, tw = Mt * NT * NBATCH;
      wmma_gemm_f32<64, false, true, true, 2><<<nblk(tw), 256, 0, stream>>>(
          X2, w2, P + AFF2, SKIP, out, 64, Mt, tw); }
}